// LABlock_49709951484790
// MI455X (gfx1250) — compile-verified
//
#include <hip/hip_runtime.h>

typedef __attribute__((ext_vector_type(2))) float v2f;
typedef __attribute__((ext_vector_type(8))) float v8f;

#define C_DIM 64
#define TILE 16
#define HS 18                    // 16 + 2 halo
#define XS_CH (HS * HS)          // 324 floats per channel
#define XS_TOT (C_DIM * XS_CH)   // 20736 floats
#define WS_TOT (5 * C_DIM * 49)  // 15680 floats
#define X1_STRIDE 272            // 256 + 16: keeps half-wave K-slices on disjoint banks

__global__ __launch_bounds__(256)
void lablock_fused(const float* __restrict__ x,
                   const float* __restrict__ w,
                   const float* __restrict__ cw,
                   const float* __restrict__ cb,
                   float* __restrict__ out)
{
    // 145,664 B static LDS; x1 (64 x 272 = 69,632 B) aliases the xs region after a barrier
    __shared__ float smem[XS_TOT + WS_TOT];
    float* xs = smem;
    float* ws = smem + XS_TOT;
    float* x1 = smem;

    const int tid = threadIdx.x;
    const int b   = blockIdx.z;
    const int h0  = blockIdx.y * TILE;
    const int w0  = blockIdx.x * TILE;

    // ---- Phase 1: load halo tile (64 x 18 x 18) and stencil table into LDS ----
    const float* xb = x + (size_t)b * C_DIM * 65536;
    for (int it = 0; it < 81; ++it) {               // 81*256 == 20736 exactly
        int i  = it * 256 + tid;
        int c  = i / XS_CH;
        int r  = i - c * XS_CH;
        int ry = r / HS;
        int rx = r - ry * HS;
        int gh = h0 - 1 + ry;
        int gw = w0 - 1 + rx;
        float v = 0.0f;
        if ((unsigned)gh < 256u && (unsigned)gw < 256u)
            v = xb[(size_t)c * 65536 + gh * 256 + gw];
        xs[i] = v;
    }
    for (int it = 0; it < 62; ++it) {
        int i = it * 256 + tid;
        if (i < WS_TOT) ws[i] = w[i];               // same flat order: [n][c][hm*7+wm]
    }
    __syncthreads();

    // ---- Phase 2: 5-point position-weighted stencil + channel softmax (thread = pixel) ----
    const int py  = tid >> 4;
    const int px  = tid & 15;
    const int pos = ((h0 + py) % 7) * 7 + ((w0 + px) % 7);

    float att[C_DIM];
    float ctr[C_DIM];
    float mx = -1e30f;
    #pragma unroll
    for (int c = 0; c < C_DIM; ++c) {
        const float* xc = xs + c * XS_CH + (py + 1) * HS + (px + 1);
        const float* wc = ws + c * 49 + pos;
        float c0 = xc[0];
        float a  = wc[0]        * c0;        // (0,0)
        a += wc[1 * 3136] * xc[-HS];         // (-1,0)
        a += wc[2 * 3136] * xc[ HS];         // (+1,0)
        a += wc[3 * 3136] * xc[-1];          // (0,-1)
        a += wc[4 * 3136] * xc[ 1];          // (0,+1)
        att[c] = a;
        ctr[c] = c0;
        mx = fmaxf(mx, a);
    }
    float s = 0.0f;
    #pragma unroll
    for (int c = 0; c < C_DIM; ++c) {
        float e = __expf(att[c] - mx);
        att[c] = e;
        s += e;
    }
    float inv = 1.0f / s;

    __syncthreads();   // all xs/ws reads done before x1 overwrites the region

    #pragma unroll
    for (int c = 0; c < C_DIM; ++c)
        x1[c * X1_STRIDE + tid] = ctr[c] * (1.0f + att[c] * inv);
    __syncthreads();

    // ---- Phase 3: 1x1 conv = GEMM via v_wmma_f32_16x16x4_f32 ----
    const int wave = tid >> 5;
    const int lane = tid & 31;
    const int o0   = (wave & 3) * 16;       // o-tile for this wave
    const int ptb  = (wave >> 2) * 8;       // first pixel-row tile
    const int mrow = lane & 15;
    const int kb   = (lane < 16) ? 0 : 2;   // K sub-slice per ISA 16x4 fp32 A layout

    v2f afrag[16];
    #pragma unroll
    for (int k0 = 0; k0 < 16; ++k0)
        afrag[k0] = *(const v2f*)(cw + (o0 + mrow) * 64 + 4 * k0 + kb);

    const int oofs = o0 + ((lane < 16) ? 0 : 8);
    v8f bias;
    #pragma unroll
    for (int j = 0; j < 8; ++j)
        bias[j] = cb[oofs + j];             // C/D layout: VGPR j -> M = j (+8 upper half)

    float* outb = out + (size_t)b * C_DIM * 65536;
    #pragma unroll
    for (int t = 0; t < 8; ++t) {
        int pt = ptb + t;                   // pixel row h0+pt, cols w0..w0+15
        const float* bp = x1 + pt * 16 + mrow;
        v8f acc = bias;
        #pragma unroll
        for (int k0 = 0; k0 < 16; ++k0) {
            v2f bfrag;
            bfrag.x = bp[(4 * k0 + kb)     * X1_STRIDE];
            bfrag.y = bp[(4 * k0 + kb + 1) * X1_STRIDE];
            acc = __builtin_amdgcn_wmma_f32_16x16x4_f32(
                false, afrag[k0], false, bfrag, (short)0, acc, false, false);
        }
        size_t base = (size_t)(h0 + pt) * 256 + (w0 + mrow);
        #pragma unroll
        for (int j = 0; j < 8; ++j)
            outb[(size_t)(oofs + j) * 65536 + base] = acc[j];
    }
}

extern "C" void kernel_launch(void* const* d_in, const int* in_sizes, int n_in,
                              void* d_out, int out_size, void* d_ws, size_t ws_size,
                              hipStream_t stream) {
    const float* x  = (const float*)d_in[0];   // [16, 64, 256, 256]
    const float* w  = (const float*)d_in[1];   // [5, 64, 7, 7]
    const float* cw = (const float*)d_in[2];   // [64, 64]
    const float* cb = (const float*)d_in[3];   // [64]
    float* out = (float*)d_out;                // [16, 64, 256, 256]
    dim3 grid(16, 16, 16);                     // (w-tile, h-tile, batch)
    lablock_fused<<<grid, 256, 0, stream>>>(x, w, cw, cb, out);
}